// APPNP_ModelBen2_45792941310034
// MI455X (gfx1250) — compile-verified
//
#include <hip/hip_runtime.h>

#define N_NODES 100000
#define N_EDGES 1600000
#define IN_DIM  128
#define NHID    64
#define OUT_DIM 40
#define K_ITERS 10
#define ALPHA   0.1f

typedef __attribute__((ext_vector_type(2))) float v2f;
typedef __attribute__((ext_vector_type(8))) float v8f;

// ---------------- degree / normalization ----------------
__global__ void deg_init_kernel(float* deg, int n) {
  int i = blockIdx.x * blockDim.x + threadIdx.x;
  if (i < n) deg[i] = 1.0f;  // self-loop contribution
}

__global__ void deg_accum_kernel(const int* __restrict__ col, float* deg, int e) {
  int i = blockIdx.x * blockDim.x + threadIdx.x;
  if (i < e) atomicAdd(&deg[col[i]], 1.0f);
}

__global__ void dinv_kernel(float* deg, int n) {
  int i = blockIdx.x * blockDim.x + threadIdx.x;
  if (i < n) { float d = deg[i]; deg[i] = (d > 0.0f) ? rsqrtf(d) : 0.0f; }
}

__global__ void norm_kernel(const int* __restrict__ row, const int* __restrict__ col,
                            const float* __restrict__ dinv, float* __restrict__ norm, int e) {
  int i = blockIdx.x * blockDim.x + threadIdx.x;
  if (i < e) norm[i] = dinv[row[i]] * dinv[col[i]];
}

__global__ void zero_kernel(float* p, int n) {
  int i = blockIdx.x * blockDim.x + threadIdx.x;
  if (i < n) p[i] = 0.0f;
}

// ---------------- APPNP propagation ----------------
// One thread = one edge x 4 consecutive feature dims (float4 gather, 4 atomics).
template <int D>
__global__ void scatter_kernel(const int* __restrict__ row, const int* __restrict__ col,
                               const float* __restrict__ norm, const float* __restrict__ h,
                               float* __restrict__ agg, int total4) {
  int idx = blockIdx.x * blockDim.x + threadIdx.x;
  if (idx >= total4) return;
  constexpr int Q = D / 4;
  int e  = idx / Q;
  int d0 = (idx - e * Q) * 4;
  if (d0 == 0 && e + 8192 < N_EDGES) {   // stream-ahead prefetch of edge data
    __builtin_prefetch(row + e + 8192, 0, 1);
    __builtin_prefetch(col + e + 8192, 0, 1);
    __builtin_prefetch(norm + e + 8192, 0, 1);
  }
  int r = row[e], c = col[e];
  float nv = norm[e];
  const float4 hv = *reinterpret_cast<const float4*>(h + (size_t)r * D + d0);
  float* ap = agg + (size_t)c * D + d0;
  atomicAdd(ap + 0, nv * hv.x);
  atomicAdd(ap + 1, nv * hv.y);
  atomicAdd(ap + 2, nv * hv.z);
  atomicAdd(ap + 3, nv * hv.w);
}

// out = (1-a) * (agg + dinv[i]^2 * h) + a * h0   (self-loop folded densely),
// then re-zero agg for the next iteration (fused zeroing pass).
template <int D>
__global__ void axpy_kernel(float* __restrict__ agg, const float* __restrict__ h,
                            const float* __restrict__ h0, const float* __restrict__ dinv,
                            float* __restrict__ out, int total4) {
  int idx = blockIdx.x * blockDim.x + threadIdx.x;
  if (idx >= total4) return;
  constexpr int Q = D / 4;
  int i = idx / Q;
  float di = dinv[i];
  float dd = di * di;
  float4* aggv = reinterpret_cast<float4*>(agg);
  const float4 a   = aggv[idx];
  const float4 hv  = reinterpret_cast<const float4*>(h)[idx];
  const float4 h0v = reinterpret_cast<const float4*>(h0)[idx];
  float4 o;
  o.x = (1.0f - ALPHA) * (a.x + dd * hv.x) + ALPHA * h0v.x;
  o.y = (1.0f - ALPHA) * (a.y + dd * hv.y) + ALPHA * h0v.y;
  o.z = (1.0f - ALPHA) * (a.z + dd * hv.z) + ALPHA * h0v.z;
  o.w = (1.0f - ALPHA) * (a.w + dd * hv.w) + ALPHA * h0v.w;
  reinterpret_cast<float4*>(out)[idx] = o;
  aggv[idx] = make_float4(0.0f, 0.0f, 0.0f, 0.0f);
}

// ---------------- WMMA f32 GEMM: Y[n x OD] = X[n x KD] * W[OD x KD]^T + bias ----------------
// Block = CTN*RTN waves. Block covers RTN*16 rows x CTN*16 cols; each wave one 16x16 tile.
template <int KD, int OD, int CTN, int RTN, bool RELU>
__global__ __launch_bounds__(CTN * RTN * 32)
void gemm_wmma_kernel(const float* __restrict__ X, const float* __restrict__ W,
                      const float* __restrict__ bias, float* __restrict__ Y, int n) {
  constexpr int CTR  = CTN * 16;     // padded output columns
  constexpr int ROWS = RTN * 16;     // rows per block
  constexpr int LDW  = KD + 4;       // LDS stride padding (kills stride-KD bank conflicts)
  constexpr int NT   = CTN * RTN * 32;
  __shared__ float sW[CTR * LDW];
  __shared__ float sX[ROWS * LDW];

  const int tid   = threadIdx.x;
  const int rows0 = blockIdx.x * ROWS;

  // Stage W (zero-padded tile rows) into LDS, float4 granularity
  for (int i4 = tid; i4 < CTR * KD / 4; i4 += NT) {
    int lin = i4 * 4;
    int r = lin / KD, k = lin - r * KD;
    float4 v = make_float4(0.0f, 0.0f, 0.0f, 0.0f);
    if (r < OD) v = *reinterpret_cast<const float4*>(W + r * KD + k);
    sW[r * LDW + k + 0] = v.x; sW[r * LDW + k + 1] = v.y;
    sW[r * LDW + k + 2] = v.z; sW[r * LDW + k + 3] = v.w;
  }
  // Stage X row-tile into LDS (fused ReLU for layer 2), float4 granularity
  for (int i4 = tid; i4 < ROWS * KD / 4; i4 += NT) {
    int lin = i4 * 4;
    int r = lin / KD, k = lin - r * KD;
    int g = rows0 + r;
    float4 v = make_float4(0.0f, 0.0f, 0.0f, 0.0f);
    if (g < n) v = *reinterpret_cast<const float4*>(X + (size_t)g * KD + k);
    if (RELU) {
      v.x = v.x > 0.0f ? v.x : 0.0f; v.y = v.y > 0.0f ? v.y : 0.0f;
      v.z = v.z > 0.0f ? v.z : 0.0f; v.w = v.w > 0.0f ? v.w : 0.0f;
    }
    sX[r * LDW + k + 0] = v.x; sX[r * LDW + k + 1] = v.y;
    sX[r * LDW + k + 2] = v.z; sX[r * LDW + k + 3] = v.w;
  }
  __syncthreads();

  const int wave = tid >> 5;
  const int lane = tid & 31;
  const int lh   = lane & 15;   // M (A) / N (B,C,D) index
  const int hi   = lane >> 4;   // half-wave selector
  const int rt   = wave / CTN;
  const int ct   = wave - rt * CTN;
  const int col0 = ct * 16;
  const int rowt = rt * 16;

  v8f acc = {};
  const float* xr = &sX[(rowt + lh) * LDW];
  const float* wr = &sW[(col0 + lh) * LDW];

  // ISA layout 16x16x4 f32:
  //  A: vgpr0 = A[M=lh][k+2*hi],   vgpr1 = A[M=lh][k+2*hi+1]
  //  B: vgpr0 = B[k+2*hi][N=lh],   vgpr1 = B[k+2*hi+1][N=lh]  (B[k][n] = W[n][k])
  for (int k = 0; k < KD; k += 4) {
    v2f a, b;
    a.x = xr[k + 2 * hi];
    a.y = xr[k + 2 * hi + 1];
    b.x = wr[k + 2 * hi];
    b.y = wr[k + 2 * hi + 1];
    acc = __builtin_amdgcn_wmma_f32_16x16x4_f32(false, a, false, b, (short)0, acc,
                                                false, false);
  }

  const int col = col0 + lh;
  const float bv = (col < OD) ? bias[col] : 0.0f;
#pragma unroll
  for (int v = 0; v < 8; ++v) {
    int r = rows0 + rowt + v + 8 * hi;   // C/D: vgpr v -> row v + 8*hi
    if (r < n && col < OD) Y[(size_t)r * OD + col] = acc[v] + bv;
  }
}

// ---------------- host launch ----------------
extern "C" void kernel_launch(void* const* d_in, const int* in_sizes, int n_in,
                              void* d_out, int out_size, void* d_ws, size_t ws_size,
                              hipStream_t stream) {
  (void)in_sizes; (void)n_in; (void)out_size; (void)ws_size;
  const float* x  = (const float*)d_in[0];
  const int*   ei = (const int*)d_in[1];
  const float* W1 = (const float*)d_in[2];
  const float* b1 = (const float*)d_in[3];
  const float* W2 = (const float*)d_in[4];
  const float* b2 = (const float*)d_in[5];
  const int* row = ei;             // edge_index[0] = sources
  const int* col = ei + N_EDGES;   // edge_index[1] = targets

  float* ws   = (float*)d_ws;
  float* dinv = ws;                           // N
  float* norm = dinv + N_NODES;               // E
  float* h0   = norm + N_EDGES;               // N*64
  float* hX   = h0 + (size_t)N_NODES * NHID;  // N*64
  float* hY   = hX + (size_t)N_NODES * NHID;  // N*64
  float* agg  = hY + (size_t)N_NODES * NHID;  // N*64 (shared by both layers)
  // layer-2 aliases (layer-1 buffers dead by then; 40 <= 64)
  float* g0 = h0;
  float* gX = hX;
  float* gY = hY;

  const int T = 256;
  auto blocks = [](int total, int bs) { return (total + bs - 1) / bs; };

  // ---- normalization ----
  deg_init_kernel<<<blocks(N_NODES, T), T, 0, stream>>>(dinv, N_NODES);
  deg_accum_kernel<<<blocks(N_EDGES, T), T, 0, stream>>>(col, dinv, N_EDGES);
  dinv_kernel<<<blocks(N_NODES, T), T, 0, stream>>>(dinv, N_NODES);
  norm_kernel<<<blocks(N_EDGES, T), T, 0, stream>>>(row, col, dinv, norm, N_EDGES);

  // ---- layer 1: h0 = X*W1^T + b1 ----  (32 rows/block, 8 waves)
  const int gemmBlocks = (N_NODES + 31) / 32;  // 3125
  gemm_wmma_kernel<IN_DIM, NHID, 4, 2, false>
      <<<gemmBlocks, 8 * 32, 0, stream>>>(x, W1, b1, h0, N_NODES);

  // one explicit zero per call (handles harness poison); axpy re-zeroes thereafter
  zero_kernel<<<blocks(N_NODES * NHID, T), T, 0, stream>>>(agg, N_NODES * NHID);

  // ---- layer 1 propagation (K=10, d=64) ----
  {
    const int nd4 = N_NODES * NHID / 4;
    const int ed4 = N_EDGES * NHID / 4;
    const float* cur = h0;
    for (int k = 0; k < K_ITERS; ++k) {
      float* nxt = (k % 2 == 0) ? hX : hY;
      scatter_kernel<NHID><<<blocks(ed4, T), T, 0, stream>>>(row, col, norm, cur, agg, ed4);
      axpy_kernel<NHID><<<blocks(nd4, T), T, 0, stream>>>(agg, cur, h0, dinv, nxt, nd4);
      cur = nxt;
    }
    // cur == hY after 10 iterations

    // ---- layer 2: g0 = relu(h)*W2^T + b2 ----  (32 rows/block, 6 waves)
    gemm_wmma_kernel<NHID, OUT_DIM, 3, 2, true>
        <<<gemmBlocks, 6 * 32, 0, stream>>>(cur, W2, b2, g0, N_NODES);
  }

  // ---- layer 2 propagation (K=10, d=40), last step writes d_out ----
  // agg[0 .. N*40) is zero here: layer-1's final axpy re-zeroed the whole N*64 region.
  {
    const int nd4 = N_NODES * OUT_DIM / 4;
    const int ed4 = N_EDGES * OUT_DIM / 4;
    const float* cur = g0;
    for (int k = 0; k < K_ITERS; ++k) {
      float* nxt = (k == K_ITERS - 1) ? (float*)d_out : ((k % 2 == 0) ? gX : gY);
      scatter_kernel<OUT_DIM><<<blocks(ed4, T), T, 0, stream>>>(row, col, norm, cur, agg, ed4);
      axpy_kernel<OUT_DIM><<<blocks(nd4, T), T, 0, stream>>>(agg, cur, g0, dinv, nxt, nd4);
      cur = nxt;
    }
  }
}